// GCN_5686536700269
// MI455X (gfx1250) — compile-verified
//
#include <hip/hip_runtime.h>
#include <hip/hip_bf16.h>

typedef __attribute__((ext_vector_type(2))) float v2f;
typedef __attribute__((ext_vector_type(8))) float v8f;

// ---------------------------------------------------------------- utilities

__global__ __launch_bounds__(256) void zero_kernel(float* __restrict__ p, long long n) {
    long long i = (long long)blockIdx.x * blockDim.x + threadIdx.x;
    long long stride = (long long)gridDim.x * blockDim.x;
    for (; i < n; i += stride) p[i] = 0.0f;
}

__global__ __launch_bounds__(256) void degree_kernel(const int* __restrict__ dst,
                                                     float* __restrict__ deg, int E) {
    int i = blockIdx.x * blockDim.x + threadIdx.x;
    if (i < E) atomicAdd(&deg[dst[i]], 1.0f);
}

__global__ __launch_bounds__(256) void dinv_kernel(const float* __restrict__ deg,
                                                   float* __restrict__ dinv, int n) {
    int i = blockIdx.x * blockDim.x + threadIdx.x;
    if (i < n) dinv[i] = rsqrtf(deg[i] + 1.0f);
}

// ------------------------------------------------- fp32 WMMA GEMM (16x16 tiles)
// C[M x NC_REAL] = A[M x K] @ B[K x NC_REAL] (+bias). NC = NC_REAL padded to 16.
// One wave computes one 16x16 tile via V_WMMA_F32_16X16X4_F32, K-step 4.
// A frag (16x4 f32): lanes 0-15 -> M=lane, V0=K, V1=K+1 ; lanes 16-31 -> K+2,K+3.
// C/D (16x16 f32): VGPR r, lanes 0-15 -> M=r, N=lane ; lanes 16-31 -> M=r+8.
template <int K, int NC, bool ADD_BIAS, int NC_REAL>
__global__ __launch_bounds__(256) void wmma_gemm(const float* __restrict__ A,
                                                 const float* __restrict__ B,
                                                 const float* __restrict__ bias,
                                                 float* __restrict__ C, int M) {
    const int wave = (blockIdx.x * blockDim.x + threadIdx.x) >> 5;
    const int lane = threadIdx.x & 31;
    constexpr int colTiles = NC / 16;
    const int rowTile = wave / colTiles;
    const int colTile = wave - rowTile * colTiles;
    const int row0 = rowTile * 16;
    if (row0 >= M) return;
    const int col0 = colTile * 16;
    const int half = lane >> 4;     // 0: K,K+1   1: K+2,K+3
    const int l16  = lane & 15;

    const float* arow = A + (size_t)(row0 + l16) * K;
    v8f acc = {};
#pragma unroll
    for (int k = 0; k < K; k += 4) {
        const int ka = k + 2 * half;
        v2f a;
        a.x = arow[ka];
        a.y = arow[ka + 1];
        v2f b;
        if constexpr (NC == NC_REAL) {
            b.x = B[(size_t)ka * NC_REAL + col0 + l16];
            b.y = B[(size_t)(ka + 1) * NC_REAL + col0 + l16];
        } else {
            const int c = col0 + l16;
            b.x = (c < NC_REAL) ? B[(size_t)ka * NC_REAL + c] : 0.0f;
            b.y = (c < NC_REAL) ? B[(size_t)(ka + 1) * NC_REAL + c] : 0.0f;
        }
        acc = __builtin_amdgcn_wmma_f32_16x16x4_f32(
            /*neg_a=*/false, a, /*neg_b=*/false, b,
            /*c_mod=*/(short)0, acc, /*reuse_a=*/false, /*reuse_b=*/false);
    }

#pragma unroll
    for (int r = 0; r < 8; ++r) {
        const int row = row0 + r + 8 * half;
        const int col = col0 + l16;
        if constexpr (NC == NC_REAL) {
            float v = acc[r];
            if constexpr (ADD_BIAS) v += bias[col];
            C[(size_t)row * NC_REAL + col] = v;
        } else {
            if (col < NC_REAL) {
                float v = acc[r];
                if constexpr (ADD_BIAS) v += bias[col];
                C[(size_t)row * NC_REAL + col] = v;
            }
        }
    }
}

// ----------------------------------------------------- edge scatter-aggregate
// agg[dst] += h[src] * dinv[src]*dinv[dst], one thread per (edge, float4 chunk)
template <int F>
__global__ __launch_bounds__(256) void aggregate_kernel(const float* __restrict__ h,
                                                        const int* __restrict__ src,
                                                        const int* __restrict__ dst,
                                                        const float* __restrict__ dinv,
                                                        float* __restrict__ agg, int E) {
    constexpr int CH = F / 4;
    const int idx = blockIdx.x * blockDim.x + threadIdx.x;
    if (idx >= E * CH) return;
    const int e = idx / CH;
    const int c = idx - e * CH;
    const int s = src[e];
    const int d = dst[e];
    const float nrm = dinv[s] * dinv[d];
    const float4 v = ((const float4*)h)[(size_t)s * CH + c];
    float* o = agg + (size_t)d * F + c * 4;
    atomicAdd(o + 0, v.x * nrm);
    atomicAdd(o + 1, v.y * nrm);
    atomicAdd(o + 2, v.z * nrm);
    atomicAdd(o + 3, v.w * nrm);
}

// --------------------------------------- out = relu(agg + hlin*dinv^2 + bias)
template <int F>
__global__ __launch_bounds__(256) void combine_kernel(const float* __restrict__ hlin,
                                                      const float* __restrict__ aggv,
                                                      const float* __restrict__ dinv,
                                                      const float* __restrict__ bias,
                                                      float* __restrict__ out, int Nn) {
    const int idx = blockIdx.x * blockDim.x + threadIdx.x;
    if (idx >= Nn * F) return;
    const int node = idx / F;
    const int f = idx - node * F;
    const float di = dinv[node];
    const float v = aggv[idx] + hlin[idx] * (di * di) + bias[f];
    out[idx] = v > 0.0f ? v : 0.0f;
}

// --------------------------------------------------------------------- driver

extern "C" void kernel_launch(void* const* d_in, const int* in_sizes, int n_in,
                              void* d_out, int out_size, void* d_ws, size_t ws_size,
                              hipStream_t stream) {
    const float* x   = (const float*)d_in[0];
    const int*   src = (const int*)  d_in[1];
    const int*   dst = (const int*)  d_in[2];
    const float* W1  = (const float*)d_in[3];
    const float* b1  = (const float*)d_in[4];
    const float* W2  = (const float*)d_in[5];
    const float* b2  = (const float*)d_in[6];
    const float* Wfc = (const float*)d_in[7];
    const float* bfc = (const float*)d_in[8];

    const int N = in_sizes[0] / 128;   // 100000
    const int E = in_sizes[1];         // 1600000

    float* out    = (float*)d_out;
    float* emb    = out;                       // [N,32]
    float* logits = out + (size_t)N * 32;      // [N,10]

    // workspace layout (floats): deg[N] | dinv[N] | buf1[N*64] | buf2[N*64]
    float* ws    = (float*)d_ws;
    float* deg   = ws;
    float* dinv  = ws + N;
    float* buf1  = ws + 2 * (size_t)N;             // h1lin; later h2lin + agg2
    float* buf2  = buf1 + (size_t)N * 64;          // agg1 -> h1 (in place)
    float* h2lin = buf1;                           // [N,32]
    float* agg2  = buf1 + (size_t)N * 32;          // [N,32]

    const int TB = 256;

    // degree + normalization
    zero_kernel<<<1024, TB, 0, stream>>>(deg, N);
    zero_kernel<<<4096, TB, 0, stream>>>(buf2, (long long)N * 64);   // agg1 = 0
    degree_kernel<<<(E + TB - 1) / TB, TB, 0, stream>>>(dst, deg, E);
    dinv_kernel<<<(N + TB - 1) / TB, TB, 0, stream>>>(deg, dinv, N);

    // layer 1: h1lin = x @ W1
    {
        const int waves  = ((N + 15) / 16) * (64 / 16);
        const int blocks = (waves * 32 + TB - 1) / TB;
        wmma_gemm<128, 64, false, 64><<<blocks, TB, 0, stream>>>(x, W1, nullptr, buf1, N);
    }
    aggregate_kernel<64><<<(E * 16 + TB - 1) / TB, TB, 0, stream>>>(buf1, src, dst, dinv, buf2, E);
    combine_kernel<64><<<(N * 64 + TB - 1) / TB, TB, 0, stream>>>(buf1, buf2, dinv, b1, buf2, N);

    // layer 2: h2lin = h1 @ W2
    zero_kernel<<<2048, TB, 0, stream>>>(agg2, (long long)N * 32);
    {
        const int waves  = ((N + 15) / 16) * (32 / 16);
        const int blocks = (waves * 32 + TB - 1) / TB;
        wmma_gemm<64, 32, false, 32><<<blocks, TB, 0, stream>>>(buf2, W2, nullptr, h2lin, N);
    }
    aggregate_kernel<32><<<(E * 8 + TB - 1) / TB, TB, 0, stream>>>(h2lin, src, dst, dinv, agg2, E);
    combine_kernel<32><<<(N * 32 + TB - 1) / TB, TB, 0, stream>>>(h2lin, agg2, dinv, b2, emb, N);

    // head: logits = emb @ Wfc + bfc   (NC padded 10 -> 16)
    {
        const int waves  = (N + 15) / 16;
        const int blocks = (waves * 32 + TB - 1) / TB;
        wmma_gemm<32, 16, true, 10><<<blocks, TB, 0, stream>>>(emb, Wfc, bfc, logits, N);
    }
}